// LuongAttention_90958817395434
// MI455X (gfx1250) — compile-verified
//
#include <hip/hip_runtime.h>
#include <hip/hip_bf16.h>

#define B_   32
#define T_   2048
#define HENC 1024
#define HDEC 1024
#define NTC  16          // T-chunks for streaming kernels
#define TCR  (T_ / NTC)  // 128 rows per chunk

typedef __attribute__((ext_vector_type(2))) float v2f;
typedef __attribute__((ext_vector_type(4))) float v4f;
typedef __attribute__((ext_vector_type(8))) float v8f;

// ---------------------------------------------------------------------------
// Kernel A: qt[b,e] = sum_h query[b,h] * W[h,e]   (32x1024x1024 GEMM)
// One wave per 16x16 output tile, V_WMMA_F32_16X16X4_F32, K stepped by 4.
// A-frag layout (ISA 7.12.2, 32-bit A 16x4): lanes 0-15 hold K=0,1; 16-31 K=2,3.
// C/D layout: VGPR r -> M = r + 8*(lane>>4), N = lane&15.
// ---------------------------------------------------------------------------
__global__ __launch_bounds__(32) void qt_gemm_wmma(const float* __restrict__ query,
                                                   const float* __restrict__ W,
                                                   float* __restrict__ qt) {
    const int lane  = threadIdx.x & 31;
    const int tile  = blockIdx.x;      // 0..127
    const int mt    = tile >> 6;       // 0..1   (M tiles: 32 rows)
    const int nt    = tile & 63;       // 0..63  (N tiles: 1024 cols)
    const int r16   = lane & 15;
    const int khalf = lane >> 4;       // 0 or 1
    const int arow  = mt * 16 + r16;   // batch row
    const int bcol  = nt * 16 + r16;   // output column e

    v8f acc = {0.f, 0.f, 0.f, 0.f, 0.f, 0.f, 0.f, 0.f};
    const float* qrow = query + (size_t)arow * HDEC;
    for (int k = 0; k < HDEC; k += 4) {
        const int kk = k + 2 * khalf;
        v2f a, b;
        a.x = qrow[kk];
        a.y = qrow[kk + 1];
        b.x = W[(size_t)kk * HENC + bcol];
        b.y = W[(size_t)(kk + 1) * HENC + bcol];
        acc = __builtin_amdgcn_wmma_f32_16x16x4_f32(false, a, false, b,
                                                    (short)0, acc, false, false);
    }
#pragma unroll
    for (int r = 0; r < 8; ++r) {
        const int m = mt * 16 + r + 8 * khalf;
        qt[(size_t)m * HENC + bcol] = acc[r];
    }
}

// ---------------------------------------------------------------------------
// Kernel B: scores[b,t] = mask ? (keys[b,t,:] . qt[b,:]) * scale : -inf
// grid = (NTC t-chunks, 32 batches), 256 threads (8 waves). Wave per row,
// non-temporal 128-bit key streaming (keys are single-use: keep L2 for the
// hot qt/scores buffers), qt staged in LDS, wave32 shuffle reduction.
// ---------------------------------------------------------------------------
__global__ __launch_bounds__(256) void scores_kernel(const float* __restrict__ keys,
                                                     const unsigned char* __restrict__ mask,
                                                     const float* __restrict__ qt,
                                                     const float* __restrict__ scale,
                                                     float* __restrict__ scores) {
    __shared__ float qs[HENC];
    const int b   = blockIdx.y;
    const int tc  = blockIdx.x;
    const int tid = threadIdx.x;
    for (int i = tid; i < HENC; i += 256) qs[i] = qt[(size_t)b * HENC + i];
    __syncthreads();

    const float sc   = scale[0];
    const int  wave  = tid >> 5;
    const int  lane  = tid & 31;
    const int  t0    = tc * TCR;
    const float* kb  = keys + (size_t)b * T_ * HENC;

    for (int i = 0; i < TCR / 8; ++i) {
        const int t = t0 + wave + 8 * i;
        const float* krow = kb + (size_t)t * HENC;
        float sum = 0.f;
#pragma unroll
        for (int j = 0; j < 8; ++j) {
            const int e = 4 * (j * 32 + lane);
            const v4f kv = __builtin_nontemporal_load((const v4f*)(krow + e));
            sum += kv.x * qs[e] + kv.y * qs[e + 1] + kv.z * qs[e + 2] + kv.w * qs[e + 3];
        }
#pragma unroll
        for (int off = 16; off > 0; off >>= 1) sum += __shfl_xor(sum, off, 32);
        if (lane == 0) {
            const float s = mask[(size_t)b * T_ + t] ? sum * sc : -__builtin_inff();
            scores[(size_t)b * T_ + t] = s;
        }
    }
}

// ---------------------------------------------------------------------------
// Kernel C: softmax over T per batch. 32 blocks x 256 threads, 8 elems/thread.
// Writes alphas directly into d_out's alphas region.
// ---------------------------------------------------------------------------
__global__ __launch_bounds__(256) void softmax_kernel(const float* __restrict__ scores,
                                                      float* __restrict__ alphas) {
    __shared__ float red[8];
    const int b    = blockIdx.x;
    const int tid  = threadIdx.x;
    const int wave = tid >> 5;
    const int lane = tid & 31;
    const float* srow = scores + (size_t)b * T_;

    float vals[8];
    float vmax = -__builtin_inff();
#pragma unroll
    for (int i = 0; i < 8; ++i) {
        vals[i] = srow[tid + 256 * i];
        vmax = fmaxf(vmax, vals[i]);
    }
#pragma unroll
    for (int off = 16; off > 0; off >>= 1) vmax = fmaxf(vmax, __shfl_xor(vmax, off, 32));
    if (lane == 0) red[wave] = vmax;
    __syncthreads();
    float gmax = red[0];
#pragma unroll
    for (int i = 1; i < 8; ++i) gmax = fmaxf(gmax, red[i]);
    __syncthreads();

    float lsum = 0.f;
#pragma unroll
    for (int i = 0; i < 8; ++i) {
        vals[i] = __expf(vals[i] - gmax);   // exp(-inf - m) == 0 for masked slots
        lsum += vals[i];
    }
#pragma unroll
    for (int off = 16; off > 0; off >>= 1) lsum += __shfl_xor(lsum, off, 32);
    if (lane == 0) red[wave] = lsum;
    __syncthreads();
    float gsum = 0.f;
#pragma unroll
    for (int i = 0; i < 8; ++i) gsum += red[i];
    const float inv = 1.0f / gsum;
#pragma unroll
    for (int i = 0; i < 8; ++i) alphas[(size_t)b * T_ + tid + 256 * i] = vals[i] * inv;
}

// ---------------------------------------------------------------------------
// Kernel D: partial context. grid = (NTC t-chunks, 32 batches), 256 threads,
// 4 e-values per thread. Non-temporal streaming of values (single-use);
// skips alpha==0 (masked) rows — a uniform branch saving ~10% of traffic.
// ---------------------------------------------------------------------------
__global__ __launch_bounds__(256) void context_partial(const float* __restrict__ values,
                                                       const float* __restrict__ alphas,
                                                       float* __restrict__ partial) {
    __shared__ float al[TCR];
    const int b   = blockIdx.y;
    const int tc  = blockIdx.x;
    const int tid = threadIdx.x;
    if (tid < TCR) al[tid] = alphas[(size_t)b * T_ + tc * TCR + tid];
    __syncthreads();

    const float* vb = values + (size_t)b * T_ * HENC + (size_t)tc * TCR * HENC;
    float acc0 = 0.f, acc1 = 0.f, acc2 = 0.f, acc3 = 0.f;
    for (int t = 0; t < TCR; ++t) {
        const float a = al[t];
        if (a == 0.f) continue;            // masked (or underflowed) rows: adds 0
        const float* vrow = vb + (size_t)t * HENC;
        acc0 += a * __builtin_nontemporal_load(vrow + tid);
        acc1 += a * __builtin_nontemporal_load(vrow + tid + 256);
        acc2 += a * __builtin_nontemporal_load(vrow + tid + 512);
        acc3 += a * __builtin_nontemporal_load(vrow + tid + 768);
    }
    float* p = partial + ((size_t)b * NTC + tc) * HENC;
    p[tid]       = acc0;
    p[tid + 256] = acc1;
    p[tid + 512] = acc2;
    p[tid + 768] = acc3;
}

// ---------------------------------------------------------------------------
// Kernel E: deterministic reduce of the NTC partials -> context in d_out.
// ---------------------------------------------------------------------------
__global__ __launch_bounds__(256) void context_reduce(const float* __restrict__ partial,
                                                      float* __restrict__ context) {
    const int b   = blockIdx.x;
    const int tid = threadIdx.x;
#pragma unroll
    for (int q = 0; q < 4; ++q) {
        const int e = tid + 256 * q;
        float s = 0.f;
#pragma unroll
        for (int tc = 0; tc < NTC; ++tc)
            s += partial[((size_t)b * NTC + tc) * HENC + e];
        context[(size_t)b * HENC + e] = s;
    }
}

extern "C" void kernel_launch(void* const* d_in, const int* in_sizes, int n_in,
                              void* d_out, int out_size, void* d_ws, size_t ws_size,
                              hipStream_t stream) {
    const float*         query  = (const float*)d_in[0];         // [32,1,1024]
    const unsigned char* mask   = (const unsigned char*)d_in[1]; // [32,2048] bool (1 byte)
    const float*         values = (const float*)d_in[2];         // [32,2048,1024]
    const float*         keys   = (const float*)d_in[3];         // [32,2048,1024]
    const float*         W      = (const float*)d_in[4];         // [1024,1024]
    const float*         scale  = (const float*)d_in[5];         // [1]

    float* out     = (float*)d_out;
    float* context = out;                  // B*HENC   = 32768 floats
    float* alphas  = out + B_ * HENC;      // B*T      = 65536 floats

    float* ws      = (float*)d_ws;
    float* qt      = ws;                   // 32768 floats
    float* scores  = ws + 32768;           // 65536 floats
    float* partial = ws + 32768 + 65536;   // 32*NTC*1024 = 524288 floats (2 MB)

    // A: project the query through W (instead of projecting 64M key elements)
    qt_gemm_wmma<<<128, 32, 0, stream>>>(query, W, qt);

    // B: bilinear scores, masked (streams keys once, NT hint)
    dim3 gS(NTC, 32);
    scores_kernel<<<gS, 256, 0, stream>>>(keys, mask, qt, scale, scores);

    // C: softmax -> alphas (directly into d_out)
    softmax_kernel<<<32, 256, 0, stream>>>(scores, alphas);

    // D: partial weighted sums of values (streams values once, NT hint)
    dim3 gC(NTC, 32);
    context_partial<<<gC, 256, 0, stream>>>(values, alphas, partial);

    // E: deterministic final reduction -> context
    context_reduce<<<32, 256, 0, stream>>>(partial, context);
}